// StateVector_2448131359097
// MI455X (gfx1250) — compile-verified
//
#include <hip/hip_runtime.h>

// ---------------------------------------------------------------------------
// StateVector pipeline for MI455X (gfx1250, wave32).
//   out[b] = |WHT( P( exp(-i ang/2) .* WHT(state[b]/||state||) ) )|^2
// All WHT-256 sub-transforms are done as  H16 * M * H16  with f32 WMMA
// (16x16x4, K chained x4), H16 entries generated once per thread from
// popcount (+-1 exact; symmetric, so one set serves both multiplies).
// Scaling (2^-8 per WHT) and 1/||state||^2 folded into the final |.|^2.
// Streamed-once input/output use non-temporal hints so the 128 MiB ping-pong
// intermediates stay resident in the 192 MB L2.
// ws layout: [1024 f32 partials][buf0 re|im planes 64MiB][buf1 re|im 64MiB]
// ---------------------------------------------------------------------------

namespace {
constexpr int   kNState      = 1 << 16;   // 65536
constexpr int   kBatch       = 128;
constexpr int   kRBlk        = 32;        // 256-element segments per workgroup
constexpr int   kTPad        = 260;       // padded LDS tile row stride (floats)
constexpr int   kThreads     = 256;       // 8 waves of 32
constexpr int   kWaves       = kThreads / 32;
constexpr int   kRowsPerWave = kRBlk / kWaves;  // 4
constexpr float kInv2p32     = 2.3283064365386963e-10f;  // 2^-32
}

typedef float v2f __attribute__((ext_vector_type(2)));
typedef float v4f __attribute__((ext_vector_type(4)));
typedef float v8f __attribute__((ext_vector_type(8)));

__device__ __forceinline__ float hsgn(int i, int j) {
  return (__builtin_popcount(i & j) & 1) ? -1.0f : 1.0f;
}

// Per-thread loop-invariant H16 operand slices. hsgn is symmetric, so the
// B-operand slices of (M*H) and the A-operand slices of (H*P) are the same
// 4 register pairs: hs[k] = { H[4k+2hi][m], H[4k+2hi+1][m] }.
__device__ __forceinline__ void make_hslices(int lane, v2f hs[4]) {
  const int m  = lane & 15;
  const int hi = lane >> 4;
#pragma unroll
  for (int k = 0; k < 4; ++k) {
    const int ca = 4 * k + 2 * hi;
    hs[k].x = hsgn(ca, m);
    hs[k].y = hsgn(ca + 1, m);
  }
}

// Unscaled 256-point WHT of one natural-order row in LDS: Y = H16 * M * H16.
// 8x v_wmma_f32_16x16x4_f32 per call. scr = 256-float per-wave LDS scratch.
// Per ISA 7.12.2: A 16x4 f32 -> lanes 0-15 row M=lane {K0,K1}, lanes 16-31
// {K2,K3}; B 4x16 -> VGPR j lane L: row 4k + (L<16 ? j : 2+j), col L&15;
// C/D 16x16 -> VGPR v lane L: (M = v + 8*(L>>4), N = L&15).
__device__ __forceinline__ v8f wht256_wave(const float* row, float* scr,
                                           int lane, const v2f hs[4]) {
  const int m  = lane & 15;
  const int hi = lane >> 4;
  v8f p = {0.f, 0.f, 0.f, 0.f, 0.f, 0.f, 0.f, 0.f};
#pragma unroll
  for (int k = 0; k < 4; ++k) {
    const int ca = 4 * k + 2 * hi;
    v2f a;
    a.x = row[m * 16 + ca];
    a.y = row[m * 16 + ca + 1];
    p = __builtin_amdgcn_wmma_f32_16x16x4_f32(false, a, false, hs[k], (short)0,
                                              p, false, false);
  }
  // P (D-layout) -> scratch in natural order; same-wave LDS ops are in-order.
#pragma unroll
  for (int v = 0; v < 8; ++v) scr[(v + 8 * hi) * 16 + m] = p[v];
  v8f y = {0.f, 0.f, 0.f, 0.f, 0.f, 0.f, 0.f, 0.f};
#pragma unroll
  for (int k = 0; k < 4; ++k) {
    const int ra = 4 * k + 2 * hi;
    v2f bm;
    bm.x = scr[ra * 16 + m];
    bm.y = scr[(ra + 1) * 16 + m];
    y = __builtin_amdgcn_wmma_f32_16x16x4_f32(false, hs[k], false, bm, (short)0,
                                              y, false, false);
  }
  return y;
}

// Pass 1: WHT along low 8 index bits; transposed store T[c*256+r]; sumsq.
__global__ __launch_bounds__(kThreads)
void k_wht_lo(const float* __restrict__ gRe, const float* __restrict__ gIm,
              float* __restrict__ tRe, float* __restrict__ tIm,
              float* __restrict__ partial) {
  __shared__ float sRe[kRBlk][kTPad];
  __shared__ float sIm[kRBlk][kTPad];
  __shared__ float sScr[kWaves * 256];
  __shared__ float sRed[kThreads];

  const int    b    = blockIdx.y;
  const int    r0   = blockIdx.x * kRBlk;
  const int    tid  = threadIdx.x;
  const size_t base = (size_t)b * kNState + (size_t)r0 * 256;

  float acc = 0.f;
#pragma unroll
  for (int i = 0; i < 8; ++i) {
    const int q   = tid + kThreads * i;  // float4 index 0..2047
    const int row = q >> 6;
    const int c4  = (q & 63) << 2;
    // Inputs are consumed exactly once -> non-temporal (keep L2 for buffers).
    const v4f vr =
        __builtin_nontemporal_load(reinterpret_cast<const v4f*>(gRe + base) + q);
    const v4f vi =
        __builtin_nontemporal_load(reinterpret_cast<const v4f*>(gIm + base) + q);
    *reinterpret_cast<v4f*>(&sRe[row][c4]) = vr;
    *reinterpret_cast<v4f*>(&sIm[row][c4]) = vi;
    acc += vr.x * vr.x + vr.y * vr.y + vr.z * vr.z + vr.w * vr.w;
    acc += vi.x * vi.x + vi.y * vi.y + vi.z * vi.z + vi.w * vi.w;
  }
  sRed[tid] = acc;
  __syncthreads();
  for (int s = kThreads / 2; s > 0; s >>= 1) {
    if (tid < s) sRed[tid] += sRed[tid + s];
    __syncthreads();
  }
  if (tid == 0) partial[b * 8 + blockIdx.x] = sRed[0];

  const int wave = tid >> 5, lane = tid & 31;
  float*    scr  = sScr + wave * 256;
  v2f       hs[4];
  make_hslices(lane, hs);
  for (int t = 0; t < kRowsPerWave; ++t) {
    const int rr  = wave * kRowsPerWave + t;
    const v8f yre = wht256_wave(&sRe[rr][0], scr, lane, hs);
    const v8f yim = wht256_wave(&sIm[rr][0], scr, lane, hs);
    const int m = lane & 15, hi = lane >> 4;
#pragma unroll
    for (int v = 0; v < 8; ++v) {
      const int c = (v + 8 * hi) * 16 + m;
      sRe[rr][c] = yre[v];
      sIm[rr][c] = yim[v];
    }
  }
  __syncthreads();

  float* oRe = tRe + (size_t)b * kNState;
  float* oIm = tIm + (size_t)b * kNState;
  for (int i = 0; i < kRBlk; ++i) {    // 128B runs per wave-store
    const int c = wave * kRBlk + i;
    oRe[(size_t)c * 256 + r0 + lane] = sRe[lane][c];
    oIm[(size_t)c * 256 + r0 + lane] = sIm[lane][c];
  }
}

// Pass 2: WHT along high 8 bits (rows of T are fixed-c) + phase, in place.
// Phase split: exp(-i(angR+angC)/2) = exp(-i angR/2) * exp(-i angC/2);
// the r-part is a 256-entry (cos,sin) LDS table (1 sincos per thread), the
// c-part is one sincos per row.
__global__ __launch_bounds__(kThreads)
void k_wht_hi_phase(float* __restrict__ tRe, float* __restrict__ tIm,
                    const float* __restrict__ thetas) {
  __shared__ float sRe[kRBlk][kTPad];
  __shared__ float sIm[kRBlk][kTPad];
  __shared__ float sScr[kWaves * 256];
  __shared__ float sTh[16];
  __shared__ float sAngC[256];
  __shared__ float sCosR[256];
  __shared__ float sSinR[256];

  const int    b    = blockIdx.y;
  const int    c0   = blockIdx.x * kRBlk;
  const int    tid  = threadIdx.x;
  const size_t base = (size_t)b * kNState + (size_t)c0 * 256;

  if (tid < 16) sTh[tid] = thetas[b * 16 + tid];
#pragma unroll
  for (int i = 0; i < 8; ++i) {
    const int q   = tid + kThreads * i;
    const int row = q >> 6;
    const int c4  = (q & 63) << 2;
    *reinterpret_cast<v4f*>(&sRe[row][c4]) =
        reinterpret_cast<const v4f*>(tRe + base)[q];
    *reinterpret_cast<v4f*>(&sIm[row][c4]) =
        reinterpret_cast<const v4f*>(tIm + base)[q];
  }
  __syncthreads();
  {  // ang(idx) = angR(idx>>8) + angC(idx&255); bit p of idx <-> theta[15-p]
    float aC = 0.f, aR = 0.f;
#pragma unroll
    for (int p = 0; p < 8; ++p) {
      const float bit = (float)((tid >> p) & 1);
      aC += bit * sTh[15 - p];
      aR += bit * sTh[7 - p];
    }
    sAngC[tid] = aC;
    float sr, cr;
    __sincosf(-0.5f * aR, &sr, &cr);
    sCosR[tid] = cr;
    sSinR[tid] = sr;
  }
  __syncthreads();

  const int wave = tid >> 5, lane = tid & 31;
  float*    scr  = sScr + wave * 256;
  v2f       hs[4];
  make_hslices(lane, hs);
  for (int t = 0; t < kRowsPerWave; ++t) {
    const int rr  = wave * kRowsPerWave + t;
    const v8f yre = wht256_wave(&sRe[rr][0], scr, lane, hs);
    const v8f yim = wht256_wave(&sIm[rr][0], scr, lane, hs);
    const int m = lane & 15, hi = lane >> 4;
    float     sc, cc;
    __sincosf(-0.5f * sAngC[c0 + rr], &sc, &cc);  // c-part phase of this row
#pragma unroll
    for (int v = 0; v < 8; ++v) {
      const int   r  = (v + 8 * hi) * 16 + m;
      const float pc = sCosR[r] * cc - sSinR[r] * sc;  // combined cos
      const float ps = sSinR[r] * cc + sCosR[r] * sc;  // combined sin
      const float re = yre[v], im = yim[v];
      sRe[rr][r] = re * pc - im * ps;
      sIm[rr][r] = im * pc + re * ps;
    }
  }
  __syncthreads();
#pragma unroll
  for (int i = 0; i < 8; ++i) {
    const int q   = tid + kThreads * i;
    const int row = q >> 6;
    const int c4  = (q & 63) << 2;
    reinterpret_cast<v4f*>(tRe + base)[q] = *reinterpret_cast<v4f*>(&sRe[row][c4]);
    reinterpret_cast<v4f*>(tIm + base)[q] = *reinterpret_cast<v4f*>(&sIm[row][c4]);
  }
}

// Pass 3: gather via cnot_p (source stored c-major) + WHT along low 8 bits;
// transposed store to buf1[c*256+r].
__global__ __launch_bounds__(kThreads)
void k_perm_wht_lo(const float* __restrict__ uRe, const float* __restrict__ uIm,
                   const int* __restrict__ cnot, float* __restrict__ tRe,
                   float* __restrict__ tIm) {
  __shared__ float sRe[kRBlk][kTPad];
  __shared__ float sIm[kRBlk][kTPad];
  __shared__ float sScr[kWaves * 256];

  const int    b   = blockIdx.y;
  const int    r0  = blockIdx.x * kRBlk;
  const int    tid = threadIdx.x;
  const float* gRe = uRe + (size_t)b * kNState;
  const float* gIm = uIm + (size_t)b * kNState;

  for (int i = 0; i < kRBlk; ++i) {
    const int p   = cnot[(r0 + i) * 256 + tid];       // coalesced index read
    const int src = ((p & 255) << 8) | (p >> 8);      // natural -> c-major
    sRe[i][tid] = gRe[src];                           // L2-resident gather
    sIm[i][tid] = gIm[src];
  }
  __syncthreads();

  const int wave = tid >> 5, lane = tid & 31;
  float*    scr  = sScr + wave * 256;
  v2f       hs[4];
  make_hslices(lane, hs);
  for (int t = 0; t < kRowsPerWave; ++t) {
    const int rr  = wave * kRowsPerWave + t;
    const v8f yre = wht256_wave(&sRe[rr][0], scr, lane, hs);
    const v8f yim = wht256_wave(&sIm[rr][0], scr, lane, hs);
    const int m = lane & 15, hi = lane >> 4;
#pragma unroll
    for (int v = 0; v < 8; ++v) {
      const int c = (v + 8 * hi) * 16 + m;
      sRe[rr][c] = yre[v];
      sIm[rr][c] = yim[v];
    }
  }
  __syncthreads();

  float* oRe = tRe + (size_t)b * kNState;
  float* oIm = tIm + (size_t)b * kNState;
  for (int i = 0; i < kRBlk; ++i) {
    const int c = wave * kRBlk + i;
    oRe[(size_t)c * 256 + r0 + lane] = sRe[lane][c];
    oIm[(size_t)c * 256 + r0 + lane] = sIm[lane][c];
  }
}

// Pass 4: WHT along high 8 bits + scaled |.|^2 -> natural-order output.
__global__ __launch_bounds__(kThreads)
void k_wht_hi_prob(const float* __restrict__ tRe, const float* __restrict__ tIm,
                   const float* __restrict__ partial, float* __restrict__ out) {
  __shared__ float sRe[kRBlk][kTPad];
  __shared__ float sIm[kRBlk][kTPad];
  __shared__ float sScr[kWaves * 256];

  const int    b    = blockIdx.y;
  const int    c0   = blockIdx.x * kRBlk;
  const int    tid  = threadIdx.x;
  const size_t base = (size_t)b * kNState + (size_t)c0 * 256;

  float ssq = 0.f;
#pragma unroll
  for (int k = 0; k < 8; ++k) ssq += partial[b * 8 + k];
  const float scale = kInv2p32 / ssq;  // 2^-8 amplitude per WHT, 1/norm^2

#pragma unroll
  for (int i = 0; i < 8; ++i) {
    const int q   = tid + kThreads * i;
    const int row = q >> 6;
    const int c4  = (q & 63) << 2;
    *reinterpret_cast<v4f*>(&sRe[row][c4]) =
        reinterpret_cast<const v4f*>(tRe + base)[q];
    *reinterpret_cast<v4f*>(&sIm[row][c4]) =
        reinterpret_cast<const v4f*>(tIm + base)[q];
  }
  __syncthreads();

  const int wave = tid >> 5, lane = tid & 31;
  float*    scr  = sScr + wave * 256;
  v2f       hs[4];
  make_hslices(lane, hs);
  for (int t = 0; t < kRowsPerWave; ++t) {
    const int rr  = wave * kRowsPerWave + t;
    const v8f yre = wht256_wave(&sRe[rr][0], scr, lane, hs);
    const v8f yim = wht256_wave(&sIm[rr][0], scr, lane, hs);
    const int m = lane & 15, hi = lane >> 4;
#pragma unroll
    for (int v = 0; v < 8; ++v) {
      const int r = (v + 8 * hi) * 16 + m;
      sRe[rr][r]  = (yre[v] * yre[v] + yim[v] * yim[v]) * scale;
    }
  }
  __syncthreads();

  float* o = out + (size_t)b * kNState;
  for (int i = 0; i < kRBlk; ++i) {  // natural order, 128B runs per wave
    const int r = wave * kRBlk + i;
    // Output is written once and never re-read -> non-temporal store.
    __builtin_nontemporal_store(sRe[lane][r], o + (size_t)r * 256 + c0 + lane);
  }
}

extern "C" void kernel_launch(void* const* d_in, const int* in_sizes, int n_in,
                              void* d_out, int out_size, void* d_ws,
                              size_t ws_size, hipStream_t stream) {
  const float* stateRe = (const float*)d_in[0];
  const float* stateIm = (const float*)d_in[1];
  const float* thetas  = (const float*)d_in[2];
  const int*   cnot    = (const int*)d_in[3];
  float*       out     = (float*)d_out;

  // ws: 1024 partials, then 4 planes of 128*65536 f32 (128 MiB + 4 KiB total)
  float* partial = (float*)d_ws;
  float* buf0Re  = partial + 1024;
  float* buf0Im  = buf0Re + (size_t)kBatch * kNState;
  float* buf1Re  = buf0Im + (size_t)kBatch * kNState;
  float* buf1Im  = buf1Re + (size_t)kBatch * kNState;

  const dim3 grid(kNState / 256 / kRBlk, kBatch);  // (8, 128)
  const dim3 block(kThreads);

  hipLaunchKernelGGL(k_wht_lo, grid, block, 0, stream, stateRe, stateIm, buf0Re,
                     buf0Im, partial);
  hipLaunchKernelGGL(k_wht_hi_phase, grid, block, 0, stream, buf0Re, buf0Im,
                     thetas);
  hipLaunchKernelGGL(k_perm_wht_lo, grid, block, 0, stream, buf0Re, buf0Im,
                     cnot, buf1Re, buf1Im);
  hipLaunchKernelGGL(k_wht_hi_prob, grid, block, 0, stream, buf1Re, buf1Im,
                     partial, out);

  (void)in_sizes; (void)n_in; (void)out_size; (void)ws_size;
}